// GCA_83167746720143
// MI455X (gfx1250) — compile-verified
//
#include <hip/hip_runtime.h>

typedef __attribute__((ext_vector_type(2))) float v2f;
typedef __attribute__((ext_vector_type(8))) float v8f;

#define CH 256   // IN == HID == PROJ == 256

__device__ inline void fma4(float4& a, const float4 v, const float w) {
    a.x = fmaf(v.x, w, a.x);
    a.y = fmaf(v.y, w, a.y);
    a.z = fmaf(v.z, w, a.z);
    a.w = fmaf(v.w, w, a.w);
}

// ---------------------------------------------------------------------------
// CSR construction: degree histogram -> exclusive scan -> bin src ids by dst
// ---------------------------------------------------------------------------
__global__ void k_zero_int(int* __restrict__ p, int n) {
    int i = blockIdx.x * blockDim.x + threadIdx.x;
    if (i < n) p[i] = 0;
}

__global__ void k_deg_count(const int* __restrict__ dst, int* __restrict__ cnt, int E) {
    int i = blockIdx.x * blockDim.x + threadIdx.x;
    if (i < E) atomicAdd(&cnt[dst[i]], 1);
}

// single-block exclusive scan over cnt[0..N) -> rowptr/cursor; fuses dinv.
__global__ void k_scan(const int* __restrict__ cnt, int* __restrict__ rowptr,
                       int* __restrict__ cursor, float* __restrict__ dinv, int N) {
    __shared__ int sh[1024];
    __shared__ int carry_s;
    if (threadIdx.x == 0) carry_s = 0;
    __syncthreads();
    for (int base = 0; base < N; base += 1024) {
        const int i = base + (int)threadIdx.x;
        const int v = (i < N) ? cnt[i] : 0;
        const int c = carry_s;
        sh[threadIdx.x] = v;
        __syncthreads();
        #pragma unroll
        for (int off = 1; off < 1024; off <<= 1) {
            const int t = (threadIdx.x >= (unsigned)off) ? sh[threadIdx.x - off] : 0;
            __syncthreads();
            sh[threadIdx.x] += t;
            __syncthreads();
        }
        const int incl = sh[threadIdx.x];
        if (i < N) {
            const int r = c + incl - v;             // exclusive
            rowptr[i] = r;
            cursor[i] = r;
            dinv[i]   = rsqrtf((float)(v + 1));     // degree incl. self-loop
        }
        __syncthreads();
        if (threadIdx.x == 1023) carry_s = c + sh[1023];
        __syncthreads();
    }
    if (threadIdx.x == 0) rowptr[N] = carry_s;
}

__global__ void k_bin(const int* __restrict__ src, const int* __restrict__ dst,
                      int* __restrict__ cursor, int* __restrict__ ebuf, int E) {
    int e = blockIdx.x * blockDim.x + threadIdx.x;
    if (e < E) {
        const int pos = atomicAdd(&cursor[dst[e]], 1);
        ebuf[pos] = src[e];
    }
}

// ---------------------------------------------------------------------------
// Dense GEMM  out[N,256] = A[N,256] @ W[256,256]  via V_WMMA_F32_16X16X4_F32
// One wave computes a 16x64 output strip (4 n-tiles share each A fragment).
// A layout (16x4 f32): lanes 0-15 hold K=k,k+1 ; lanes 16-31 hold K=k+2,k+3.
// C/D (16x16 f32): vgpr v holds row M=v (lanes 0-15) / M=v+8 (lanes 16-31).
// ---------------------------------------------------------------------------
__global__ void k_gemm_wmma(const float* __restrict__ A, const float* __restrict__ W,
                            float* __restrict__ out, int N) {
    const int wave = (blockIdx.x * blockDim.x + threadIdx.x) >> 5;
    const int lane = threadIdx.x & 31;
    const int mtiles = (N + 15) >> 4;
    if (wave >= mtiles * 4) return;                 // wave-uniform: EXEC all-1s

    const int m0    = (wave >> 2) << 4;
    const int nbase = (wave & 3) << 6;              // 64-column strip
    const bool hi = lane >= 16;
    const int  l  = lane & 15;

    int arow = m0 + l;
    if (arow >= N) arow = N - 1;                    // clamp (duplicate read only)
    const float* __restrict__ Ar = A + (size_t)arow * CH;

    v8f c0 = {}, c1 = {}, c2 = {}, c3 = {};
    #pragma unroll 2
    for (int k = 0; k < CH; k += 4) {
        const int ka = k + (hi ? 2 : 0);
        v2f av;  av.x = Ar[ka];  av.y = Ar[ka + 1];
        const float* __restrict__ Wr0 = W + (size_t)ka * CH + nbase + l;
        const float* __restrict__ Wr1 = Wr0 + CH;
        v2f b0 = {Wr0[0],  Wr1[0]};
        v2f b1 = {Wr0[16], Wr1[16]};
        v2f b2 = {Wr0[32], Wr1[32]};
        v2f b3 = {Wr0[48], Wr1[48]};
        c0 = __builtin_amdgcn_wmma_f32_16x16x4_f32(false, av, false, b0, (short)0, c0, false, false);
        c1 = __builtin_amdgcn_wmma_f32_16x16x4_f32(false, av, false, b1, (short)0, c1, false, false);
        c2 = __builtin_amdgcn_wmma_f32_16x16x4_f32(false, av, false, b2, (short)0, c2, false, false);
        c3 = __builtin_amdgcn_wmma_f32_16x16x4_f32(false, av, false, b3, (short)0, c3, false, false);
    }

    const int mb = m0 + (hi ? 8 : 0);
    if (m0 + 16 <= N) {                             // full tile: branch-free stores
        #pragma unroll
        for (int i = 0; i < 8; ++i) {
            float* __restrict__ o = out + (size_t)(mb + i) * CH + nbase + l;
            o[0]  = c0[i];
            o[16] = c1[i];
            o[32] = c2[i];
            o[48] = c3[i];
        }
    } else {
        #pragma unroll
        for (int i = 0; i < 8; ++i) {
            const int m = mb + i;
            if (m < N) {
                float* __restrict__ o = out + (size_t)m * CH + nbase + l;
                o[0]  = c0[i];
                o[16] = c1[i];
                o[32] = c2[i];
                o[48] = c3[i];
            }
        }
    }
}

// ---------------------------------------------------------------------------
// CSR gather aggregation, fused bias + PReLU. One wave per destination node;
// each lane owns 8 channels (2 x float4). No float atomics.
// out[i,:] = prelu( dinv_i^2 * h[i,:] + sum_j dinv_sj*dinv_i * h[sj,:] + b, a )
// ---------------------------------------------------------------------------
__global__ void k_gather(const float* __restrict__ h, const int* __restrict__ rowptr,
                         const int* __restrict__ ebuf, const float* __restrict__ dinv,
                         const float* __restrict__ bias, const float* __restrict__ a_ptr,
                         float* __restrict__ out, int N) {
    const int node = (blockIdx.x * blockDim.x + threadIdx.x) >> 5;
    const int lane = threadIdx.x & 31;
    if (node >= N) return;

    const float4* __restrict__ h4 = (const float4*)h;
    const int ca = lane;
    const int cb = lane + 32;

    const float di = dinv[node];
    const float self = di * di;

    float4 v0 = h4[(size_t)node * 64 + ca];
    float4 v1 = h4[(size_t)node * 64 + cb];
    float4 acc0 = make_float4(v0.x * self, v0.y * self, v0.z * self, v0.w * self);
    float4 acc1 = make_float4(v1.x * self, v1.y * self, v1.z * self, v1.w * self);

    const int beg = rowptr[node];
    const int end = rowptr[node + 1];
    for (int j = beg; j < end; ++j) {
        const int s = ebuf[j];
        const float w = dinv[s] * di;
        fma4(acc0, h4[(size_t)s * 64 + ca], w);
        fma4(acc1, h4[(size_t)s * 64 + cb], w);
    }

    const float alpha = *a_ptr;
    const float4* __restrict__ b4 = (const float4*)bias;
    const float4 ba = b4[ca];
    const float4 bb = b4[cb];
    float4 r0, r1;
    r0.x = acc0.x + ba.x; r0.y = acc0.y + ba.y; r0.z = acc0.z + ba.z; r0.w = acc0.w + ba.w;
    r1.x = acc1.x + bb.x; r1.y = acc1.y + bb.y; r1.z = acc1.z + bb.z; r1.w = acc1.w + bb.w;
    r0.x = r0.x >= 0.f ? r0.x : alpha * r0.x;
    r0.y = r0.y >= 0.f ? r0.y : alpha * r0.y;
    r0.z = r0.z >= 0.f ? r0.z : alpha * r0.z;
    r0.w = r0.w >= 0.f ? r0.w : alpha * r0.w;
    r1.x = r1.x >= 0.f ? r1.x : alpha * r1.x;
    r1.y = r1.y >= 0.f ? r1.y : alpha * r1.y;
    r1.z = r1.z >= 0.f ? r1.z : alpha * r1.z;
    r1.w = r1.w >= 0.f ? r1.w : alpha * r1.w;

    float4* __restrict__ o4 = (float4*)out;
    o4[(size_t)node * 64 + ca] = r0;
    o4[(size_t)node * 64 + cb] = r1;
}

// ---------------------------------------------------------------------------
extern "C" void kernel_launch(void* const* d_in, const int* in_sizes, int n_in,
                              void* d_out, int out_size, void* d_ws, size_t ws_size,
                              hipStream_t stream) {
    const float* x  = (const float*)d_in[0];
    const float* W1 = (const float*)d_in[1];
    const float* b1 = (const float*)d_in[2];
    const float* W2 = (const float*)d_in[3];
    const float* b2 = (const float*)d_in[4];
    const float* a  = (const float*)d_in[5];
    const int* eidx = (const int*)d_in[6];

    const int N = in_sizes[0] / CH;
    const int E = in_sizes[6] / 2;
    const int* src = eidx;          // edge_index[0]
    const int* dst = eidx + E;      // edge_index[1]
    float* out = (float*)d_out;

    // workspace carve-up (256B aligned slices)
    auto align256 = [](size_t b) { return (b + 255) & ~(size_t)255; };
    char* w = (char*)d_ws;
    float* dinv   = (float*)w;  w += align256((size_t)N * 4);
    int*   cnt    = (int*)w;    w += align256((size_t)N * 4);
    int*   rowptr = (int*)w;    w += align256((size_t)(N + 1) * 4);
    int*   cursor = (int*)w;    w += align256((size_t)N * 4);
    int*   ebuf   = (int*)w;    w += align256((size_t)E * 4);
    float* h      = (float*)w;  // [N*CH]

    const int T = 256;

    // ---- CSR build ----
    k_zero_int <<<(N + T - 1) / T, T, 0, stream>>>(cnt, N);
    k_deg_count<<<(E + T - 1) / T, T, 0, stream>>>(dst, cnt, E);
    k_scan     <<<1, 1024, 0, stream>>>(cnt, rowptr, cursor, dinv, N);
    k_bin      <<<(E + T - 1) / T, T, 0, stream>>>(src, dst, cursor, ebuf, E);

    const int gwaves  = ((N + 15) / 16) * 4;        // 4 waves per 16-row strip
    const int gblocks = (gwaves + 3) / 4;           // 128 threads = 4 waves
    const int ablocks = (N + 7) / 8;                // 8 waves of gather per block

    // ---- layer 1 ----
    k_gemm_wmma<<<gblocks, 128, 0, stream>>>(x, W1, h, N);
    k_gather   <<<ablocks, 256, 0, stream>>>(h, rowptr, ebuf, dinv, b1, a, out, N);

    // ---- layer 2 (reads activations from d_out; h scratch reused) ----
    k_gemm_wmma<<<gblocks, 128, 0, stream>>>(out, W2, h, N);
    k_gather   <<<ablocks, 256, 0, stream>>>(h, rowptr, ebuf, dinv, b2, a, out, N);
}